// ROIPool_5540507812230
// MI455X (gfx1250) — compile-verified
//
#include <hip/hip_runtime.h>
#include <math.h>

#define NB   4      // batch
#define NC   256    // channels
#define NH   38
#define NW   38
#define NROI 256
#define GCH  8                  // channels staged per block
#define NGRP (NC / GCH)         // 32 channel groups
#define SCALE_F 0.0625f

#if defined(__has_builtin)
#  if __has_builtin(__builtin_amdgcn_global_load_async_to_lds_b32)
#    define USE_ASYNC_LDS 1
#  else
#    define USE_ASYNC_LDS 0
#  endif
#  if __has_builtin(__builtin_amdgcn_global_load_async_to_lds_b64)
#    define USE_ASYNC_B64 1
#  else
#    define USE_ASYNC_B64 0
#  endif
#  if __has_builtin(__builtin_amdgcn_s_wait_asynccnt)
#    define USE_WAIT_BUILTIN 1
#  else
#    define USE_WAIT_BUILTIN 0
#  endif
#else
#  define USE_ASYNC_LDS 0
#  define USE_ASYNC_B64 0
#  define USE_WAIT_BUILTIN 0
#endif

typedef int v2i __attribute__((ext_vector_type(2)));
typedef __attribute__((address_space(1))) int  g_int;
typedef __attribute__((address_space(3))) int  l_int;
typedef __attribute__((address_space(1))) v2i  g_v2i;
typedef __attribute__((address_space(3))) v2i  l_v2i;

__global__ __launch_bounds__(128)
void roipool_kernel(const float* __restrict__ feat,
                    const float* __restrict__ rois,
                    const int*   __restrict__ roib,
                    float*       __restrict__ out)
{
    // 8 channels x up-to-38 rows x 38 cols, compacted by runtime row count.
    __shared__ __align__(16) float smem[GCH * NH * NW];   // 46208 bytes

    const int bid   = blockIdx.x;
    const int n     = bid / NGRP;           // ROI index
    const int cbase = (bid % NGRP) * GCH;   // first channel of this group
    const int tid   = threadIdx.x;
    const int nthr  = blockDim.x;

    const int   b  = roib[n];
    const float x1 = floorf(rois[n * 4 + 0] * SCALE_F + 0.5f);
    const float y1 = floorf(rois[n * 4 + 1] * SCALE_F + 0.5f);
    const float x2 = floorf(rois[n * 4 + 2] * SCALE_F + 0.5f);
    const float y2 = floorf(rois[n * 4 + 3] * SCALE_F + 0.5f);
    const float roi_w = fmaxf(x2 - x1 + 1.0f, 1.0f);
    const float roi_h = fmaxf(y2 - y1 + 1.0f, 1.0f);
    const float bin_h = roi_h / 7.0f;
    const float bin_w = roi_w / 7.0f;

    // Union of all vertical bins: [hstart(ph=0), hend(ph=6))  (monotone in ph).
    const int region_hs = (int)fminf(fmaxf(y1, 0.0f), (float)NH);
    const int region_he = (int)fminf(fmaxf(ceilf(7.0f * bin_h) + y1, 0.0f), (float)NH);
    const int rh   = region_he - region_hs;
    const int rhW  = rh * NW;               // contiguous floats per channel (always even: rh*38)
    const int total = GCH * rhW;

    // Full-width rows [region_hs, region_he) are contiguous in memory.
    const float* gbase = feat
        + (((size_t)b * NC + (size_t)cbase) * NH + (size_t)region_hs) * NW;

#if USE_ASYNC_LDS && USE_ASYNC_B64
    // Pairs at even i are 8B-aligned in both global and LDS; rhW even, so a
    // pair never straddles a channel boundary. 256B moved per wave issue.
    for (int i = 2 * tid; i < total; i += 2 * nthr) {
        const int ch = i / rhW;
        const int r  = i - ch * rhW;
        const float* src = gbase + (size_t)ch * (NH * NW) + r;
        __builtin_amdgcn_global_load_async_to_lds_b64(
            (g_v2i*)(uintptr_t)(const void*)src,
            (l_v2i*)(smem + i),
            0, 0);
    }
#else
    for (int i = tid; i < total; i += nthr) {
        const int ch = i / rhW;
        const int r  = i - ch * rhW;
        const float* src = gbase + (size_t)ch * (NH * NW) + r;
#  if USE_ASYNC_LDS
        __builtin_amdgcn_global_load_async_to_lds_b32(
            (g_int*)(uintptr_t)(const void*)src,
            (l_int*)(smem + i),
            0, 0);
#  else
        smem[i] = *src;
#  endif
    }
#endif
#if USE_ASYNC_LDS
#  if USE_WAIT_BUILTIN
    __builtin_amdgcn_s_wait_asynccnt(0);
#  else
    asm volatile("s_wait_asynccnt 0" ::: "memory");
#  endif
#endif
    __syncthreads();

    // 8 channels x 49 bins = 392 outputs per block.
    for (int j = tid; j < GCH * 49; j += nthr) {
        const int ch  = j / 49;
        const int bin = j - ch * 49;
        const int ph  = bin / 7;
        const int pw  = bin - ph * 7;

        const float hs_f = fminf(fmaxf(floorf((float)ph * bin_h) + y1, 0.0f), (float)NH);
        const float he_f = fminf(fmaxf(ceilf((float)(ph + 1) * bin_h) + y1, 0.0f), (float)NH);
        const float ws_f = fminf(fmaxf(floorf((float)pw * bin_w) + x1, 0.0f), (float)NW);
        const float we_f = fminf(fmaxf(ceilf((float)(pw + 1) * bin_w) + x1, 0.0f), (float)NW);
        const int hs = (int)hs_f, he = (int)he_f;
        const int ws = (int)ws_f, we = (int)we_f;

        float m = -INFINITY;
        const float* cb = smem + ch * rhW;
        for (int h = hs; h < he; ++h) {
            const float* row = cb + (h - region_hs) * NW;
            for (int w = ws; w < we; ++w) {
                m = fmaxf(m, row[w]);
            }
        }
        if (m == -INFINITY) m = 0.0f;   // reference: isneginf -> 0

        out[(((size_t)n * NC + (size_t)(cbase + ch)) * 7 + (size_t)ph) * 7 + (size_t)pw] = m;
    }
}

extern "C" void kernel_launch(void* const* d_in, const int* in_sizes, int n_in,
                              void* d_out, int out_size, void* d_ws, size_t ws_size,
                              hipStream_t stream) {
    (void)in_sizes; (void)n_in; (void)out_size; (void)d_ws; (void)ws_size;
    const float* feat = (const float*)d_in[0];
    const float* rois = (const float*)d_in[1];
    const int*   roib = (const int*)d_in[2];
    float*       out  = (float*)d_out;

    dim3 grid(NROI * NGRP);   // 8192 blocks
    dim3 block(128);          // 4 wave32 waves
    hipLaunchKernelGGL(roipool_kernel, grid, block, 0, stream, feat, rois, roib, out);
}